// LatentExpander_66666482369092
// MI455X (gfx1250) — compile-verified
//
#include <hip/hip_runtime.h>
#include <cmath>

// LatentExpander: tokens[b,l,o] = mask(z[b,l]) * ( sum_k gelu(z[b,l]*W1[l,k]+b1[l,k]) * W2[l,o,k]
//                                                  + b2[l,o] + pos[l,o] )
// B=4096, L=16, H=1024. fp32 end-to-end using V_WMMA_F32_16X16X4_F32 (memory-bound workload,
// so full-precision WMMA is free; W2 (64MB) is L2-resident, output streamed with NT stores).

typedef __attribute__((ext_vector_type(2))) float v2f;
typedef __attribute__((ext_vector_type(8))) float v8f;

#define B_DIM 4096
#define L_DIM 16
#define H_DIM 1024
#define MB    32   // b-rows per block

__global__ __launch_bounds__(256)
void latent_expander_wmma(const float* __restrict__ z,
                          const float* __restrict__ W1,
                          const float* __restrict__ b1,
                          const float* __restrict__ W2,
                          const float* __restrict__ b2,
                          const float* __restrict__ pos,
                          float* __restrict__ out) {
  // Dynamic LDS: gelu'd h tile, paired-K layout:
  //   hP[(k>>1)*64 + r*2 + (k&1)]  for r in [0,32), k in [0,1024)  -> 128 KB
  extern __shared__ float hP[];
  __shared__ float zv[MB];

  const int tid  = threadIdx.x;
  const int lane = tid & 31;
  const int wid  = tid >> 5;          // 8 waves
  const int l    = blockIdx.y;
  const int b0   = blockIdx.x * MB;

  if (tid < MB) zv[tid] = z[(size_t)(b0 + tid) * L_DIM + l];
  __syncthreads();

  // ---- Phase 1: h = gelu_exact(z*W1 + b1), staged once into LDS ----
  const float* W1l = W1 + (size_t)l * H_DIM;
  const float* b1l = b1 + (size_t)l * H_DIM;
  const float  zr  = zv[lane];                 // row r == lane
  for (int k = wid; k < H_DIM; k += 8) {
    float x = fmaf(zr, W1l[k], b1l[k]);        // W1l[k]/b1l[k] are wave-uniform (broadcast)
    float g = 0.5f * x * (1.0f + erff(x * 0.70710678118654752f));
    hP[(k >> 1) * 64 + lane * 2 + (k & 1)] = g;   // lanes hit 32 distinct banks
  }
  __syncthreads();

  // ---- Phase 2: per-wave 16x16 output tiles, K=1024 via v_wmma_f32_16x16x4_f32 ----
  // A layout (32-bit, 16x4): lanes 0-15 hold K=k0,k0+1 ; lanes 16-31 hold K=k0+2,k0+3 ; M = lane&15
  const int half = lane >> 4;
  const int lr   = lane & 15;

  for (int nt = wid; nt < H_DIM / 16; nt += 8) {
    const int    n0  = nt * 16;
    const int    o   = n0 + lr;
    // B frag: B[k][n] = W2[l, o=n, k]; per lane a contiguous float2 at k0+2*half
    const float* w2p = W2 + ((size_t)l * H_DIM + o) * H_DIM + 2 * half;

    v8f acc0 = {};   // rows b0 + 0..15
    v8f acc1 = {};   // rows b0 + 16..31  (shares B frag -> half the W2 fetches per FLOP)
    #pragma unroll 4
    for (int k0 = 0; k0 < H_DIM; k0 += 4) {
      const v2f bfrag = *(const v2f*)(w2p + k0);
      const int abase = ((k0 >> 1) + half) * 64 + lr * 2;
      const v2f a0 = *(const v2f*)(hP + abase);        // rows 0..15
      const v2f a1 = *(const v2f*)(hP + abase + 32);   // rows 16..31
      acc0 = __builtin_amdgcn_wmma_f32_16x16x4_f32(false, a0, false, bfrag,
                                                   (short)0, acc0, false, false);
      acc1 = __builtin_amdgcn_wmma_f32_16x16x4_f32(false, a1, false, bfrag,
                                                   (short)0, acc1, false, false);
    }

    // ---- Epilogue: +b2 +pos_embed, Matryoshka zero-mask, NT streaming stores ----
    const float bias = b2[(size_t)l * H_DIM + o] + pos[(size_t)l * H_DIM + o];
    #pragma unroll
    for (int i = 0; i < 8; i++) {
      const int   r0 = half * 8 + i;                       // C/D layout: M = i (+8 for hi lanes)
      const float m0 = (zv[r0]      != 0.0f) ? 1.0f : 0.0f;
      const float m1 = (zv[16 + r0] != 0.0f) ? 1.0f : 0.0f;
      float v0 = (acc0[i] + bias) * m0;
      float v1 = (acc1[i] + bias) * m1;
      __builtin_nontemporal_store(v0, &out[((size_t)(b0 + r0)      * L_DIM + l) * H_DIM + o]);
      __builtin_nontemporal_store(v1, &out[((size_t)(b0 + 16 + r0) * L_DIM + l) * H_DIM + o]);
    }
  }
}

extern "C" void kernel_launch(void* const* d_in, const int* in_sizes, int n_in,
                              void* d_out, int out_size, void* d_ws, size_t ws_size,
                              hipStream_t stream) {
  const float* z   = (const float*)d_in[0];   // [B, L]
  const float* W1  = (const float*)d_in[1];   // [L, H]
  const float* b1  = (const float*)d_in[2];   // [L, H]
  const float* W2  = (const float*)d_in[3];   // [L, H, H] (torch [out,in] per dim)
  const float* b2  = (const float*)d_in[4];   // [L, H]
  const float* pos = (const float*)d_in[5];   // [1, L, H]
  float*       out = (float*)d_out;           // [B, L, H]

  dim3 grid(B_DIM / MB, L_DIM);               // (128, 16)
  dim3 block(256);                            // 8 wave32s
  const size_t lds_bytes = (size_t)(H_DIM / 2) * 64 * sizeof(float);  // 128 KB

  latent_expander_wmma<<<grid, block, lds_bytes, stream>>>(z, W1, b1, W2, b2, pos, out);
}